// SS2DExpert_7473243095602
// MI455X (gfx1250) — compile-verified
//
#include <hip/hip_runtime.h>
#include <hip/hip_bf16.h>

typedef __attribute__((ext_vector_type(2))) float v2f;
typedef __attribute__((ext_vector_type(8))) float v8f;

#define B_  4
#define H_  32
#define W_  32
#define L_  1024
#define DIM_ 256
#define DIN_ 256
#define E_  512
#define K_  4
#define R_  64
#define N_  64
#define CD_ 192   // R + 2N

__device__ __forceinline__ int perm_l(int k, int l) {
  switch (k) {
    case 0: return l;
    case 1: return (L_ - 1) - l;
    case 2: return ((l & 31) << 5) | (l >> 5);
    default: { int l2 = (L_ - 1) - l; return ((l2 & 31) << 5) | (l2 >> 5); }
  }
}

__device__ __forceinline__ float softplusf(float x) {
  return (x > 20.0f) ? x : log1pf(expf(x));
}

// ---------------------------------------------------------------------------
// 1) in_proj GEMM: xz[m=b*L+l][e] = sum_c x[m][c] * Win[e][c]
//    M=4096, N=512, Kdim=256.  One wave -> one 16x16 tile, f32 WMMA (K=4).
// ---------------------------------------------------------------------------
__global__ void gemm_inproj(const float* __restrict__ X,
                            const float* __restrict__ Win,
                            float* __restrict__ xz) {
  int wid  = (blockIdx.x * blockDim.x + threadIdx.x) >> 5;   // 0..8191
  int lane = threadIdx.x & 31;
  int tm = wid >> 5;           // 0..255
  int tn = wid & 31;           // 0..31
  int m0 = tm << 4, n0 = tn << 4;
  int row  = lane & 15;
  int koff = (lane >> 4) << 1; // 0 or 2
  const float* arow = X   + (size_t)(m0 + row) * 256;
  const float* brow = Win + (size_t)(n0 + row) * 256;
  v8f acc = {};
  for (int kk = 0; kk < 256; kk += 4) {
    if ((kk & 63) == 0) {
      __builtin_prefetch(arow + kk + 64, 0, 3);
      __builtin_prefetch(brow + kk + 64, 0, 3);
    }
    v2f a = *(const v2f*)(arow + kk + koff);
    v2f b = *(const v2f*)(brow + kk + koff);
    acc = __builtin_amdgcn_wmma_f32_16x16x4_f32(false, a, false, b,
                                                (short)0, acc, false, false);
  }
  int mhi = (lane >> 4) << 3;
#pragma unroll
  for (int v = 0; v < 8; ++v) {
    int m = m0 + v + mhi;
    xz[(size_t)m * E_ + n0 + row] = acc[v];
  }
}

// ---------------------------------------------------------------------------
// 2) depthwise 3x3 conv + bias + SiLU.  xc[b,l,d] layout.
// ---------------------------------------------------------------------------
__global__ void conv_silu(const float* __restrict__ xz,
                          const float* __restrict__ cw,
                          const float* __restrict__ cb,
                          float* __restrict__ xc) {
  int idx = blockIdx.x * blockDim.x + threadIdx.x;   // b*L*256 total
  if (idx >= B_ * L_ * DIN_) return;
  int d = idx & 255;
  int l = (idx >> 8) & (L_ - 1);
  int b = idx >> 18;
  int h = l >> 5, w = l & 31;
  float acc = cb[d];
#pragma unroll
  for (int ky = 0; ky < 3; ++ky) {
    int hh = h + ky - 1;
    if ((unsigned)hh >= (unsigned)H_) continue;
#pragma unroll
    for (int kx = 0; kx < 3; ++kx) {
      int ww = w + kx - 1;
      if ((unsigned)ww >= (unsigned)W_) continue;
      acc += xz[(size_t)((b << 10) + (hh << 5) + ww) * E_ + d] *
             cw[d * 9 + ky * 3 + kx];
    }
  }
  xc[idx] = acc / (1.0f + expf(-acc));   // SiLU
}

// ---------------------------------------------------------------------------
// 3) x_proj GEMM per (b,k): x_dbl[bk][l][c] = sum_d xc[b, perm_k(l), d]*xw[k,c,d]
//    M=L=1024 (64 tiles), N=192 (12 tiles), Kdim=256.
// ---------------------------------------------------------------------------
__global__ void gemm_xproj(const float* __restrict__ xc,
                           const float* __restrict__ xw,
                           float* __restrict__ xdbl) {
  int wid  = (blockIdx.x * blockDim.x + threadIdx.x) >> 5;   // 0..12287
  int lane = threadIdx.x & 31;
  int bk = wid / 768;
  int r  = wid - bk * 768;
  int tm = r / 12, tn = r - tm * 12;
  int b = bk >> 2, k = bk & 3;
  int m0 = tm << 4, n0 = tn << 4;
  int row  = lane & 15;
  int koff = (lane >> 4) << 1;
  int lp = perm_l(k, m0 + row);
  const float* arow = xc + (size_t)((b << 10) + lp) * DIN_;
  const float* brow = xw + (size_t)(k * CD_ + n0 + row) * DIN_;
  v8f acc = {};
  for (int kk = 0; kk < 256; kk += 4) {
    if ((kk & 63) == 0) __builtin_prefetch(brow + kk + 64, 0, 3);
    v2f a  = *(const v2f*)(arow + kk + koff);
    v2f bb = *(const v2f*)(brow + kk + koff);
    acc = __builtin_amdgcn_wmma_f32_16x16x4_f32(false, a, false, bb,
                                                (short)0, acc, false, false);
  }
  float* out = xdbl + (size_t)bk * L_ * CD_;
  int mhi = (lane >> 4) << 3;
#pragma unroll
  for (int v = 0; v < 8; ++v) {
    int m = m0 + v + mhi;
    out[(size_t)m * CD_ + n0 + row] = acc[v];
  }
}

// ---------------------------------------------------------------------------
// 4) dt_proj GEMM + bias + softplus: delta[bk][l][d]
//    M=L (64 tiles), N=256 (16 tiles), Kdim=R=64.
// ---------------------------------------------------------------------------
__global__ void gemm_dtproj(const float* __restrict__ xdbl,
                            const float* __restrict__ dtw,
                            const float* __restrict__ dtb,
                            float* __restrict__ delta) {
  int wid  = (blockIdx.x * blockDim.x + threadIdx.x) >> 5;   // 0..16383
  int lane = threadIdx.x & 31;
  int bk = wid >> 10;
  int r  = wid & 1023;
  int tm = r >> 4, tn = r & 15;
  int k = bk & 3;
  int m0 = tm << 4, n0 = tn << 4;
  int row  = lane & 15;
  int koff = (lane >> 4) << 1;
  const float* arow = xdbl + ((size_t)bk * L_ + m0 + row) * CD_;  // c in [0,64)
  const float* brow = dtw + (size_t)(k * DIN_ + n0 + row) * R_;
  v8f acc = {};
  for (int kk = 0; kk < 64; kk += 4) {
    v2f a  = *(const v2f*)(arow + kk + koff);
    v2f bb = *(const v2f*)(brow + kk + koff);
    acc = __builtin_amdgcn_wmma_f32_16x16x4_f32(false, a, false, bb,
                                                (short)0, acc, false, false);
  }
  float bias = dtb[k * DIN_ + n0 + row];
  int mhi = (lane >> 4) << 3;
#pragma unroll
  for (int v = 0; v < 8; ++v) {
    int m = m0 + v + mhi;
    delta[((size_t)bk * L_ + m) * DIN_ + n0 + row] = softplusf(acc[v] + bias);
  }
}

// ---------------------------------------------------------------------------
// 5) selective scan.  One block = one (b,k): 1024 threads = 32 waves.
//    Thread -> (d = tid>>2, n-quarter = tid&3): 16 states in VGPRs.
//    B/C vectors (128 f32 per step) double-buffered through LDS; one barrier
//    per step overlaps next step's fetch with current step's exp/FMA work.
//    y = sum_n h*C reassembled with two shfl_xor hops.
// ---------------------------------------------------------------------------
__global__ void scan_kernel(const float* __restrict__ delta,
                            const float* __restrict__ xdbl,
                            const float* __restrict__ xc,
                            const float* __restrict__ A_log,
                            const float* __restrict__ Dp,
                            float* __restrict__ yout) {
  int bk  = blockIdx.x;            // 0..15
  int b   = bk >> 2, k = bk & 3;
  int tid = threadIdx.x;           // 0..1023
  int d   = tid >> 2;              // 0..255
  int nq  = tid & 3;               // n in [nq*16, nq*16+16)

  float Areg[16], h[16];
  const float* al = A_log + (size_t)(k * DIN_ + d) * N_ + nq * 16;
#pragma unroll
  for (int n = 0; n < 16; ++n) { Areg[n] = -__expf(al[n]); h[n] = 0.0f; }
  float Dv = Dp[k * DIN_ + d];

  const float* dptr = delta + (size_t)bk * L_ * DIN_ + d;
  const float* src  = xdbl + (size_t)bk * L_ * CD_ + R_;  // B|C: 128 f32 per l
  const float* xcb  = xc + (size_t)(b << 10) * DIN_ + d;
  float* yo         = yout + (size_t)bk * L_ * DIN_ + d;

  __shared__ float sBC[2][128];
  if (tid < 128) sBC[0][tid] = src[tid];
  __syncthreads();

  for (int l = 0; l < L_; ++l) {
    int cb = l & 1, nb = cb ^ 1;
    if (l + 1 < L_ && tid < 128) sBC[nb][tid] = src[(size_t)(l + 1) * CD_ + tid];

    float dt = dptr[(size_t)l * DIN_];
    int lp   = perm_l(k, l);
    float u  = xcb[(size_t)lp * DIN_];
    float du = dt * u;
    const float* Bb = &sBC[cb][nq * 16];
    const float* Cb = &sBC[cb][64 + nq * 16];
    float y = 0.0f;
#pragma unroll
    for (int n = 0; n < 16; ++n) {
      h[n] = __expf(dt * Areg[n]) * h[n] + du * Bb[n];
      y += h[n] * Cb[n];
    }
    y += __shfl_xor(y, 1);
    y += __shfl_xor(y, 2);
    if (nq == 0) yo[(size_t)l * DIN_] = y + u * Dv;
    __syncthreads();
  }
}

// ---------------------------------------------------------------------------
// 6) direction combine + layernorm + SiLU gate -> gated[b,l,d]
// ---------------------------------------------------------------------------
__global__ void combine_ln_gate(const float* __restrict__ yout,
                                const float* __restrict__ xz,
                                const float* __restrict__ g,
                                const float* __restrict__ bb,
                                float* __restrict__ gated) {
  int bl = blockIdx.x;              // b*L + l
  int d  = threadIdx.x;             // 0..255
  int b  = bl >> 10, l = bl & (L_ - 1);
  int p  = ((l & 31) << 5) | (l >> 5);
  const float* y0 = yout + (size_t)((b << 2) + 0) * L_ * DIN_;
  const float* y1 = yout + (size_t)((b << 2) + 1) * L_ * DIN_;
  const float* y2 = yout + (size_t)((b << 2) + 2) * L_ * DIN_;
  const float* y3 = yout + (size_t)((b << 2) + 3) * L_ * DIN_;
  float v = y0[(size_t)l * DIN_ + d] +
            y1[(size_t)(L_ - 1 - l) * DIN_ + d] +
            y2[(size_t)p * DIN_ + d] +
            y3[(size_t)(L_ - 1 - p) * DIN_ + d];

  __shared__ float sb[256];
  sb[d] = v; __syncthreads();
  for (int s = 128; s > 0; s >>= 1) { if (d < s) sb[d] += sb[d + s]; __syncthreads(); }
  float mean = sb[0] * (1.0f / 256.0f); __syncthreads();
  float c = v - mean;
  sb[d] = c * c; __syncthreads();
  for (int s = 128; s > 0; s >>= 1) { if (d < s) sb[d] += sb[d + s]; __syncthreads(); }
  float var = sb[0] * (1.0f / 256.0f);
  float yn = c * rsqrtf(var + 1e-5f) * g[d] + bb[d];
  float z = xz[(size_t)bl * E_ + DIN_ + d];
  gated[(size_t)bl * DIN_ + d] = yn * (z / (1.0f + expf(-z)));
}

// ---------------------------------------------------------------------------
// 7) deterministic mean over L + final layernorm -> out[b,d]
// ---------------------------------------------------------------------------
__global__ void pool_final(const float* __restrict__ gated,
                           const float* __restrict__ ng,
                           const float* __restrict__ nb,
                           float* __restrict__ out) {
  int b = blockIdx.x;
  int d = threadIdx.x;
  float s = 0.0f;
  for (int l = 0; l < L_; ++l) s += gated[(size_t)((b << 10) + l) * DIN_ + d];
  float pv = s * (1.0f / (float)L_);

  __shared__ float sb[256];
  sb[d] = pv; __syncthreads();
  for (int st = 128; st > 0; st >>= 1) { if (d < st) sb[d] += sb[d + st]; __syncthreads(); }
  float mean = sb[0] * (1.0f / 256.0f); __syncthreads();
  float c = pv - mean;
  sb[d] = c * c; __syncthreads();
  for (int st = 128; st > 0; st >>= 1) { if (d < st) sb[d] += sb[d + st]; __syncthreads(); }
  float var = sb[0] * (1.0f / 256.0f);
  out[b * DIN_ + d] = c * rsqrtf(var + 1e-5f) * ng[d] + nb[d];
}

// ---------------------------------------------------------------------------
extern "C" void kernel_launch(void* const* d_in, const int* in_sizes, int n_in,
                              void* d_out, int out_size, void* d_ws, size_t ws_size,
                              hipStream_t stream) {
  const float* x        = (const float*)d_in[0];
  const float* in_w     = (const float*)d_in[1];
  const float* conv_w   = (const float*)d_in[2];
  const float* conv_b   = (const float*)d_in[3];
  const float* x_proj_w = (const float*)d_in[4];
  const float* dt_w     = (const float*)d_in[5];
  const float* dt_b     = (const float*)d_in[6];
  const float* A_log    = (const float*)d_in[7];
  const float* Dp       = (const float*)d_in[8];
  const float* out_g    = (const float*)d_in[9];
  const float* out_b    = (const float*)d_in[10];
  const float* norm_g   = (const float*)d_in[11];
  const float* norm_b   = (const float*)d_in[12];
  float* out = (float*)d_out;

  // workspace layout (f32): 60 MB total
  float* ws    = (float*)d_ws;
  float* xz    = ws;                     // 4096*512   = 2,097,152
  float* xc    = xz   + 2097152;         // 4096*256   = 1,048,576
  float* xdbl  = xc   + 1048576;         // 16*1024*192= 3,145,728
  float* delta = xdbl + 3145728;         // 16*1024*256= 4,194,304
  float* yout  = delta + 4194304;        // 16*1024*256= 4,194,304
  float* gated = yout + 4194304;         // 4096*256   = 1,048,576

  gemm_inproj<<<1024, 256, 0, stream>>>(x, in_w, xz);                 // 8192 waves
  conv_silu  <<<4096, 256, 0, stream>>>(xz, conv_w, conv_b, xc);
  gemm_xproj <<<1536, 256, 0, stream>>>(xc, x_proj_w, xdbl);          // 12288 waves
  gemm_dtproj<<<2048, 256, 0, stream>>>(xdbl, dt_w, dt_b, delta);     // 16384 waves
  scan_kernel<<<16,  1024, 0, stream>>>(delta, xdbl, xc, A_log, Dp, yout); // 512 waves
  combine_ln_gate<<<4096, 256, 0, stream>>>(yout, xz, out_g, out_b, gated);
  pool_final <<<B_,   256, 0, stream>>>(gated, norm_g, norm_b, out);
}